// SimpleGNN_28578712387660
// MI455X (gfx1250) — compile-verified
//
#include <hip/hip_runtime.h>

typedef __attribute__((ext_vector_type(2))) float v2f;
typedef __attribute__((ext_vector_type(8))) float v8f;

#define GNN_N      50000
#define GNN_E      1000000
#define GNN_INCH   128
#define GNN_HID    64
#define GNN_OUT    64
#define ROW_TILES  (GNN_N / 16)       // 3125, exact
#define GNN_EPS    1e-5f
#define GNN_SLOPE  0.1f

// ---------------------------------------------------------------------------
// fp32 WMMA: D(16x16) = A(16x4) * B(4x16) + C.  Wave32; EXEC must be all 1s.
// A frag: lane m=lane&15 holds row m, K = (lane>>4)*2 + {0,1}  -> float2 load
// B frag: lane n=lane&15 holds col n, same K pattern; B[k][n] = W[n][k]
// C/D   : vgpr v, lanes 0-15 -> (M=v,     N=lane), lanes 16-31 -> (M=v+8, N=lane-16)
// ---------------------------------------------------------------------------
__device__ __forceinline__ v8f wmma_f32x4(v2f a, v2f b, v8f c) {
  return __builtin_amdgcn_wmma_f32_16x16x4_f32(
      /*neg_a=*/false, a, /*neg_b=*/false, b,
      /*c_mod=*/(short)0, c, /*reuse_a=*/false, /*reuse_b=*/false);
}

// ---------------------------------------------------------------------------
__global__ void gnn_zero_kernel(float* __restrict__ p, int n) {
  int i = blockIdx.x * blockDim.x + threadIdx.x;
  if (i < n) p[i] = 0.0f;
}

// ---------------------------------------------------------------------------
// H[N,64] = X[N,128] @ W_in^T + b_in     (W_in is [64,128], row-major)
// ---------------------------------------------------------------------------
__global__ void gnn_fc_in_kernel(const float* __restrict__ X,
                                 const float* __restrict__ W,
                                 const float* __restrict__ bias,
                                 float* __restrict__ H) {
  const int wave = (int)((blockIdx.x * blockDim.x + threadIdx.x) >> 5);
  const int lane = (int)(threadIdx.x & 31);
  if (wave >= ROW_TILES) return;                 // wave-uniform exit, EXEC stays full
  const int row0 = wave * 16;
  const int m    = lane & 15;
  const int koff = (lane >> 4) * 2;              // 0 or 2

  v8f acc[4] = {};
  for (int k = 0; k < GNN_INCH; k += 4) {
    const v2f a = *(const v2f*)(X + (size_t)(row0 + m) * GNN_INCH + k + koff);
#pragma unroll
    for (int nt = 0; nt < 4; ++nt) {
      const v2f b = *(const v2f*)(W + (size_t)(nt * 16 + m) * GNN_INCH + k + koff);
      acc[nt] = wmma_f32x4(a, b, acc[nt]);
    }
  }
  const int half = lane >> 4;
#pragma unroll
  for (int nt = 0; nt < 4; ++nt) {
#pragma unroll
    for (int v = 0; v < 8; ++v) {
      const int r = row0 + v + 8 * half;
      const int c = nt * 16 + (lane & 15);
      H[(size_t)r * GNN_HID + c] = acc[nt][v] + bias[c];
    }
  }
}

// ---------------------------------------------------------------------------
// Scatter-mean prep: one wave per edge, 2 channels per lane.
// agg[dst] += h[src]; cnt[dst] += 1
// ---------------------------------------------------------------------------
__global__ void gnn_edge_agg_kernel(const long long* __restrict__ ei,
                                    const float* __restrict__ H,
                                    float* __restrict__ agg,
                                    float* __restrict__ cnt) {
  const long long wid  = (long long)((blockIdx.x * (size_t)blockDim.x + threadIdx.x) >> 5);
  const int       lane = (int)(threadIdx.x & 31);
  if (wid >= GNN_E) return;
  const long long src = ei[wid];
  const long long dst = ei[GNN_E + wid];
  const float* hs = H + (size_t)src * GNN_HID;
  float*       ad = agg + (size_t)dst * GNN_HID;
  atomicAdd(ad + lane,      hs[lane]);
  atomicAdd(ad + lane + 32, hs[lane + 32]);
  if (lane == 0) atomicAdd(cnt + dst, 1.0f);
}

// ---------------------------------------------------------------------------
// conv = (agg/max(cnt,1)) @ W_l^T + b_l + H @ W_r^T ; also column sum / sumsq
// ---------------------------------------------------------------------------
__global__ void gnn_conv_kernel(const float* __restrict__ agg,
                                const float* __restrict__ cnt,
                                const float* __restrict__ H,
                                const float* __restrict__ Wl,
                                const float* __restrict__ bl,
                                const float* __restrict__ Wr,
                                float* __restrict__ conv,
                                float* __restrict__ colsum,
                                float* __restrict__ colsumsq) {
  const int wave = (int)((blockIdx.x * blockDim.x + threadIdx.x) >> 5);
  const int lane = (int)(threadIdx.x & 31);
  if (wave >= ROW_TILES) return;
  const int row0 = wave * 16;
  const int m    = lane & 15;
  const int koff = (lane >> 4) * 2;

  const float crow = cnt[row0 + m];
  const float invc = 1.0f / fmaxf(crow, 1.0f);

  v8f acc[4] = {};
  for (int k = 0; k < GNN_HID; k += 4) {
    v2f a1 = *(const v2f*)(agg + (size_t)(row0 + m) * GNN_HID + k + koff);
    a1[0] *= invc; a1[1] *= invc;
    const v2f a2 = *(const v2f*)(H + (size_t)(row0 + m) * GNN_HID + k + koff);
#pragma unroll
    for (int nt = 0; nt < 4; ++nt) {
      const v2f b1 = *(const v2f*)(Wl + (size_t)(nt * 16 + m) * GNN_HID + k + koff);
      const v2f b2 = *(const v2f*)(Wr + (size_t)(nt * 16 + m) * GNN_HID + k + koff);
      acc[nt] = wmma_f32x4(a1, b1, acc[nt]);
      acc[nt] = wmma_f32x4(a2, b2, acc[nt]);
    }
  }

  const int half = lane >> 4;
#pragma unroll
  for (int nt = 0; nt < 4; ++nt) {
    const int c = nt * 16 + (lane & 15);
    const float bc = bl[c];
    float s = 0.0f, s2 = 0.0f;
#pragma unroll
    for (int v = 0; v < 8; ++v) {
      const int r = row0 + v + 8 * half;
      const float val = acc[nt][v] + bc;
      conv[(size_t)r * GNN_HID + c] = val;
      s += val;
      s2 += val * val;
    }
    // lanes l and l+16 hold the same column c -> combine, lane<16 commits
    s  += __shfl_xor(s, 16, 32);
    s2 += __shfl_xor(s2, 16, 32);
    if (half == 0) {
      atomicAdd(&colsum[c], s);
      atomicAdd(&colsumsq[c], s2);
    }
  }
}

// ---------------------------------------------------------------------------
// Fold GraphNorm into per-channel (alpha*mu, scale, shift)
// var = E[c^2] - 2*a*mu^2 + a^2*mu^2
// ---------------------------------------------------------------------------
__global__ void gnn_norm_params_kernel(const float* __restrict__ colsum,
                                       const float* __restrict__ colsumsq,
                                       const float* __restrict__ gw,
                                       const float* __restrict__ gb,
                                       const float* __restrict__ gms,
                                       float* __restrict__ params) {
  const int c = (int)threadIdx.x;
  if (c >= GNN_HID) return;
  const float invN = 1.0f / (float)GNN_N;
  const float mu  = colsum[c] * invN;
  const float ex2 = colsumsq[c] * invN;
  const float a   = gms[c];
  const float var = ex2 - 2.0f * a * mu * mu + a * a * mu * mu;
  const float rs  = rsqrtf(var + GNN_EPS);
  params[c]              = a * mu;        // center
  params[GNN_HID + c]    = rs * gw[c];    // scale
  params[2 * GNN_HID + c] = gb[c];        // shift
}

// ---------------------------------------------------------------------------
// out = LeakyReLU((conv - amu)*scale + shift) @ W_out^T + b_out
// ---------------------------------------------------------------------------
__global__ void gnn_fc_out_kernel(const float* __restrict__ conv,
                                  const float* __restrict__ params,
                                  const float* __restrict__ Wout,
                                  const float* __restrict__ bout,
                                  float* __restrict__ out) {
  const int wave = (int)((blockIdx.x * blockDim.x + threadIdx.x) >> 5);
  const int lane = (int)(threadIdx.x & 31);
  if (wave >= ROW_TILES) return;
  const int row0 = wave * 16;
  const int m    = lane & 15;
  const int koff = (lane >> 4) * 2;

  v8f acc[4] = {};
  for (int k = 0; k < GNN_HID; k += 4) {
    const int c0 = k + koff;
    const v2f raw = *(const v2f*)(conv + (size_t)(row0 + m) * GNN_HID + c0);
    float n0 = (raw[0] - params[c0])     * params[GNN_HID + c0]     + params[2 * GNN_HID + c0];
    float n1 = (raw[1] - params[c0 + 1]) * params[GNN_HID + c0 + 1] + params[2 * GNN_HID + c0 + 1];
    n0 = (n0 >= 0.0f) ? n0 : GNN_SLOPE * n0;
    n1 = (n1 >= 0.0f) ? n1 : GNN_SLOPE * n1;
    v2f a; a[0] = n0; a[1] = n1;
#pragma unroll
    for (int nt = 0; nt < 4; ++nt) {
      const v2f b = *(const v2f*)(Wout + (size_t)(nt * 16 + m) * GNN_HID + c0);
      acc[nt] = wmma_f32x4(a, b, acc[nt]);
    }
  }
  const int half = lane >> 4;
#pragma unroll
  for (int nt = 0; nt < 4; ++nt) {
#pragma unroll
    for (int v = 0; v < 8; ++v) {
      const int r = row0 + v + 8 * half;
      const int c = nt * 16 + (lane & 15);
      out[(size_t)r * GNN_OUT + c] = acc[nt][v] + bout[c];
    }
  }
}

// ---------------------------------------------------------------------------
extern "C" void kernel_launch(void* const* d_in, const int* in_sizes, int n_in,
                              void* d_out, int out_size, void* d_ws, size_t ws_size,
                              hipStream_t stream) {
  (void)in_sizes; (void)n_in; (void)out_size; (void)ws_size;

  const float*     x    = (const float*)d_in[0];
  const long long* ei   = (const long long*)d_in[1];   // int64 edge_index [2, E]
  const float*     Win  = (const float*)d_in[2];
  const float*     bin  = (const float*)d_in[3];
  const float*     Wl   = (const float*)d_in[4];
  const float*     bl   = (const float*)d_in[5];
  const float*     Wr   = (const float*)d_in[6];
  const float*     gw   = (const float*)d_in[7];
  const float*     gb   = (const float*)d_in[8];
  const float*     gms  = (const float*)d_in[9];
  const float*     Wout = (const float*)d_in[10];
  const float*     bout = (const float*)d_in[11];
  float*           out  = (float*)d_out;

  // workspace layout (floats)
  float* ws       = (float*)d_ws;
  float* H        = ws;                          // 3,200,000
  float* conv     = ws + 3200000;                // 3,200,000
  float* agg      = ws + 6400000;                // 3,200,000   (zeroed)
  float* cnt      = ws + 9600000;                // 50,000      (zeroed)
  float* colsum   = ws + 9650000;                // 64          (zeroed)
  float* colsumsq = ws + 9650064;                // 64          (zeroed)
  float* params   = ws + 9650128;                // 192

  const int zero_elems = 3200000 + 50000 + 64 + 64;  // agg..colsumsq contiguous

  const int BLK        = 256;                    // 8 waves/block
  const int wavesBlk   = BLK / 32;
  const int gemm_grid  = (ROW_TILES + wavesBlk - 1) / wavesBlk;   // 391
  const int edge_grid  = (GNN_E + wavesBlk - 1) / wavesBlk;       // 125000
  const int zero_grid  = (zero_elems + BLK - 1) / BLK;

  gnn_zero_kernel<<<zero_grid, BLK, 0, stream>>>(agg, zero_elems);
  gnn_fc_in_kernel<<<gemm_grid, BLK, 0, stream>>>(x, Win, bin, H);
  gnn_edge_agg_kernel<<<edge_grid, BLK, 0, stream>>>(ei, H, agg, cnt);
  gnn_conv_kernel<<<gemm_grid, BLK, 0, stream>>>(agg, cnt, H, Wl, bl, Wr,
                                                 conv, colsum, colsumsq);
  gnn_norm_params_kernel<<<1, 64, 0, stream>>>(colsum, colsumsq, gw, gb, gms, params);
  gnn_fc_out_kernel<<<gemm_grid, BLK, 0, stream>>>(conv, params, Wout, bout, out);
}